// Subtraction_4028679323857
// MI455X (gfx1250) — compile-verified
//
#include <hip/hip_runtime.h>
#include <stdint.h>

#define KSZ  7
#define PAD  3
#define H    56
#define W    56
#define CCH  64
#define NB   8
#define HP   (H + 2*PAD)    // 62
#define WP   (W + 2*PAD)    // 62
#define NPIX (H * W)        // 3136
#define KK   (KSZ * KSZ)    // 49
#define NPLANES (NB * CCH)  // 512
#define TPB  256

typedef float f32x4 __attribute__((ext_vector_type(4)));
typedef __attribute__((address_space(1))) int* gptr_i32;
typedef __attribute__((address_space(3))) int* lptr_i32;

__device__ __forceinline__ int reflect_idx(int u, int n) {
    // np.pad 'reflect' (no edge repeat); pad (3) < n (56) so one fold suffices
    u = (u < 0) ? -u : u;
    u = (u >= n) ? (2 * n - 2 - u) : u;
    return u;
}

__global__ __launch_bounds__(TPB) void san_subtraction_kernel(
    const float* __restrict__ x, float* __restrict__ out)
{
    __shared__ float lds[HP * WP];   // 62*62*4 = 15376 B reflect-padded plane

    const int plane = blockIdx.x;    // n*C + c
    const float* __restrict__ xp = x + (size_t)plane * NPIX;
    const int tid = threadIdx.x;

    // ---- Stage 1: gather reflect-padded plane into LDS via async global->LDS
    for (int idx = tid; idx < HP * WP; idx += TPB) {
        const int ty = idx / WP;
        const int tx = idx - ty * WP;
        const int sy = reflect_idx(ty - PAD, H);
        const int sx = reflect_idx(tx - PAD, W);
        const float* g = xp + sy * W + sx;
#if __has_builtin(__builtin_amdgcn_global_load_async_to_lds_b32)
        __builtin_amdgcn_global_load_async_to_lds_b32(
            (gptr_i32)(void*)g,
            (lptr_i32)(void*)&lds[idx],
            /*offset=*/0, /*cpol=*/0);
#else
        lds[idx] = *g;
#endif
    }
#if __has_builtin(__builtin_amdgcn_s_wait_asynccnt)
    __builtin_amdgcn_s_wait_asynccnt(0);
#else
    asm volatile("s_wait_asynccnt 0" ::: "memory");
#endif
    __syncthreads();

    // ---- Stage 2: each thread emits 4 consecutive pixels per k as one b128 NT store
    float* __restrict__ outp = out + (size_t)plane * (KK * NPIX);

    for (int q = tid; q < NPIX / 4; q += TPB) {
        const int l0 = q * 4;
        int base[4];
        float c[4];
#pragma unroll
        for (int m = 0; m < 4; ++m) {
            const int l = l0 + m;
            const int y = l / W;
            const int xx = l - y * W;
            base[m] = y * WP + xx;                     // top-left of 7x7 window
            c[m] = lds[base[m] + PAD * WP + PAD];      // center
        }
#pragma unroll
        for (int i = 0; i < KSZ; ++i) {
#pragma unroll
            for (int j = 0; j < KSZ; ++j) {
                f32x4 v;
                v.x = c[0] - lds[base[0] + i * WP + j];
                v.y = c[1] - lds[base[1] + i * WP + j];
                v.z = c[2] - lds[base[2] + i * WP + j];
                v.w = c[3] - lds[base[3] + i * WP + j];
                f32x4* dst = (f32x4*)(outp + (size_t)(i * KSZ + j) * NPIX + l0);
                __builtin_nontemporal_store(v, dst);
            }
        }
    }
}

extern "C" void kernel_launch(void* const* d_in, const int* in_sizes, int n_in,
                              void* d_out, int out_size, void* d_ws, size_t ws_size,
                              hipStream_t stream) {
    const float* x = (const float*)d_in[0];
    float* out = (float*)d_out;
    san_subtraction_kernel<<<NPLANES, TPB, 0, stream>>>(x, out);
}